// FlaxAttentionModule_68710886802170
// MI455X (gfx1250) — compile-verified
//
#include <hip/hip_runtime.h>

// ---------------------------------------------------------------------------
// Decode-step KV cache update + mask + per-row int8 quantization (MI455X).
// Memory-bound: ~1.6 GB traffic -> ~70us floor at 23.3 TB/s. No GEMM exists
// in this op, so no WMMA; the gfx1250-specific path is async global->LDS
// staging (ASYNCcnt) + wave32 shuffle reduction + non-temporal stores.
// ---------------------------------------------------------------------------

namespace {
constexpr int B = 4, S = 4096, H = 32, D = 128;
constexpr long long ROWS  = (long long)B * S * H;          // 524288 rows of 128
constexpr long long ELEMS = ROWS * D;                      // 67108864

// Output layout: concatenated flat, reference return order, all as float.
constexpr long long KEY_OUT_OFF = 0;
constexpr long long VAL_OUT_OFF = ELEMS;                   //  67108864
constexpr long long MASK_OFF    = 2 * ELEMS;               // 134217728
constexpr long long KEY_Q_OFF   = MASK_OFF + (long long)B * S;   // 134234112
constexpr long long KEY_S_OFF   = KEY_Q_OFF + ELEMS;       // 201342976
constexpr long long VAL_Q_OFF   = KEY_S_OFF + ROWS;        // 201867264
constexpr long long VAL_S_OFF   = VAL_Q_OFF + ELEMS;       // 268976128
}

typedef float fvec4 __attribute__((ext_vector_type(4)));

__global__ __launch_bounds__(256) void kv_cache_quant_kernel(
    const float* __restrict__ cached_key,
    const float* __restrict__ cached_value,
    const float* __restrict__ key_new,
    const float* __restrict__ val_new,
    const int*   __restrict__ cache_index,
    float*       __restrict__ out)
{
    __shared__ fvec4 kbuf[256];
    __shared__ fvec4 vbuf[256];

    const int tid  = threadIdx.x;
    const int lane = tid & 31;
    const int wv   = tid >> 5;
    const long long row = (long long)blockIdx.x * 8 + wv;  // one (b,s,h) row / wave
    const int cur = cache_index[0];

    // row = ((b*S)+s)*H + h ; H=32, S=4096
    const int h = (int)(row & (H - 1));
    const int s = (int)((row >> 5) & (S - 1));
    const int b = (int)(row >> 17);

    // dynamic_update_slice: the row at s==cache_index comes from the new token.
    const float* ksrc;
    const float* vsrc;
    if (s == cur) {
        const long long nr = ((long long)b * H + h) * (long long)D;
        ksrc = key_new + nr;
        vsrc = val_new + nr;
    } else {
        ksrc = cached_key   + row * D;
        vsrc = cached_value + row * D;
    }

    // Per-lane 16B async copies global -> LDS (tracked by ASYNCcnt).
    const fvec4* kp = (const fvec4*)ksrc + lane;
    const fvec4* vp = (const fvec4*)vsrc + lane;
    const unsigned kl = (unsigned)(size_t)&kbuf[tid];  // low 32b of flat LDS addr == LDS offset
    const unsigned vl = (unsigned)(size_t)&vbuf[tid];

    asm volatile(
        "global_load_async_to_lds_b128 %0, %2, off\n\t"
        "global_load_async_to_lds_b128 %1, %3, off\n\t"
        "s_wait_asynccnt 0"
        :
        : "v"(kl), "v"(vl), "v"(kp), "v"(vp)
        : "memory");

    const fvec4 kv = kbuf[tid];
    const fvec4 vv = vbuf[tid];

    // Rowwise amax over D=128: local max of 4, then 5-step wave32 butterfly.
    float km = fmaxf(fmaxf(fabsf(kv.x), fabsf(kv.y)), fmaxf(fabsf(kv.z), fabsf(kv.w)));
    float vm = fmaxf(fmaxf(fabsf(vv.x), fabsf(vv.y)), fmaxf(fabsf(vv.z), fabsf(vv.w)));
#pragma unroll
    for (int m = 16; m >= 1; m >>= 1) {
        km = fmaxf(km, __shfl_xor(km, m, 32));
        vm = fmaxf(vm, __shfl_xor(vm, m, 32));
    }
    const float ks = fmaxf(km / 127.0f, 1e-8f);
    const float vs = fmaxf(vm / 127.0f, 1e-8f);

    // q = clip(round_half_even(x/scale), -128, 127)   (rintf == RNE)
    fvec4 kq, vq;
    kq.x = fminf(fmaxf(rintf(kv.x / ks), -128.0f), 127.0f);
    kq.y = fminf(fmaxf(rintf(kv.y / ks), -128.0f), 127.0f);
    kq.z = fminf(fmaxf(rintf(kv.z / ks), -128.0f), 127.0f);
    kq.w = fminf(fmaxf(rintf(kv.w / ks), -128.0f), 127.0f);
    vq.x = fminf(fmaxf(rintf(vv.x / vs), -128.0f), 127.0f);
    vq.y = fminf(fmaxf(rintf(vv.y / vs), -128.0f), 127.0f);
    vq.z = fminf(fmaxf(rintf(vv.z / vs), -128.0f), 127.0f);
    vq.w = fminf(fmaxf(rintf(vv.w / vs), -128.0f), 127.0f);

    // Streaming, write-once outputs: non-temporal 128-bit stores.
    const long long re = row * D + (long long)lane * 4;
    __builtin_nontemporal_store(kv, (fvec4*)(out + KEY_OUT_OFF + re));
    __builtin_nontemporal_store(vv, (fvec4*)(out + VAL_OUT_OFF + re));
    __builtin_nontemporal_store(kq, (fvec4*)(out + KEY_Q_OFF  + re));
    __builtin_nontemporal_store(vq, (fvec4*)(out + VAL_Q_OFF  + re));
    if (lane == 0) {
        out[KEY_S_OFF + row] = ks;
        out[VAL_S_OFF + row] = vs;
    }
}

// attn_mask_out = (arange(S) < cur+1) & attention_mask   -> emitted as 0/1 float
__global__ __launch_bounds__(256) void mask_kernel(
    const unsigned char* __restrict__ attn_mask,   // jnp.bool_ -> 1 byte/elem
    const int*           __restrict__ cache_index,
    float*               __restrict__ out_mask)
{
    const int i = blockIdx.x * 256 + threadIdx.x;
    if (i < B * S) {
        const int cur = cache_index[0];
        const int pos = i & (S - 1);               // shape (B,1,1,S)
        out_mask[i] = (attn_mask[i] != 0 && pos < cur + 1) ? 1.0f : 0.0f;
    }
}

extern "C" void kernel_launch(void* const* d_in, const int* in_sizes, int n_in,
                              void* d_out, int out_size, void* d_ws, size_t ws_size,
                              hipStream_t stream)
{
    const float*         ck = (const float*)d_in[0];          // cached_key
    const float*         cv = (const float*)d_in[1];          // cached_value
    const float*         kn = (const float*)d_in[2];          // key (new token)
    const float*         vn = (const float*)d_in[3];          // value (new token)
    const unsigned char* am = (const unsigned char*)d_in[4];  // attention_mask (bool)
    const int*           ci = (const int*)d_in[5];            // cache_index
    float* out = (float*)d_out;

    kv_cache_quant_kernel<<<(int)(ROWS / 8), 256, 0, stream>>>(ck, cv, kn, vn, ci, out);
    mask_kernel<<<(B * S + 255) / 256, 256, 0, stream>>>(am, ci, out + MASK_OFF);

    (void)in_sizes; (void)n_in; (void)out_size; (void)d_ws; (void)ws_size;
}